// GREA_4191888081317
// MI455X (gfx1250) — compile-verified
//
#include <hip/hip_runtime.h>
#include <math.h>

#define N_NODES 131072
#define G_GRAPHS 512
#define D_EMB 300
#define H_HID 600
#define HP 608            // H padded to 38 * 16 for WMMA N-tiles
#define MTILE 128         // nodes per workgroup in the gate GEMM
#define GATE_THREADS 128  // 4 waves; each wave owns a 32x16 strip (2 accumulators)

typedef __attribute__((ext_vector_type(2))) float v2f;
typedef __attribute__((ext_vector_type(8))) float v8f;

// ---------------------------------------------------------------------------
// Kernel 0: fold BatchNorm (eval) + linear bias into per-h affine (k, c).
//   gate path:  bn(x@W1 + b1) = (x@W1)*gk + gc
//   pred path:  bn((a+b)@W1 + b1) = a@W1*kp + (b@W1*kp + cB)
// Padded region [600,608) zeroed so WMMA tiles contribute exactly 0.
// ---------------------------------------------------------------------------
__global__ void grea_setup_params(
    const float* __restrict__ gb1, const float* __restrict__ gsc,
    const float* __restrict__ gbi, const float* __restrict__ gme,
    const float* __restrict__ gva, const float* __restrict__ gw2in,
    const float* __restrict__ pb1, const float* __restrict__ psc,
    const float* __restrict__ pbi, const float* __restrict__ pme,
    const float* __restrict__ pva,
    float* __restrict__ gk, float* __restrict__ gc, float* __restrict__ gw2,
    float* __restrict__ kp, float* __restrict__ cB)
{
    int h = blockIdx.x * blockDim.x + threadIdx.x;
    if (h < H_HID) {
        float kg = gsc[h] * rsqrtf(gva[h] + 1e-5f);
        gk[h] = kg;
        gc[h] = (gb1[h] - gme[h]) * kg + gbi[h];
        gw2[h] = gw2in[h];
        float kpv = psc[h] * rsqrtf(pva[h] + 1e-5f);
        kp[h] = kpv;
        cB[h] = (pb1[h] - pme[h]) * kpv + pbi[h];
    } else if (h < HP) {
        gk[h] = 0.f; gc[h] = 0.f; gw2[h] = 0.f;
    }
}

// ---------------------------------------------------------------------------
// Kernel 1: copy gate_w1 [300,600] -> padded [300,608] (zeros beyond 600)
// ---------------------------------------------------------------------------
__global__ void grea_pad_w1(const float* __restrict__ w1, float* __restrict__ w1p)
{
    int idx = blockIdx.x * blockDim.x + threadIdx.x;
    if (idx >= D_EMB * HP) return;
    int n = idx % HP, d = idx / HP;
    w1p[idx] = (n < H_HID) ? w1[d * H_HID + n] : 0.f;
}

// ---------------------------------------------------------------------------
// Kernel 2: gate = sigmoid( relu(affine(x @ W1)) . w2 + b2 ), fully fused.
// 4 waves/block; each wave owns a 32(M)x16(N) strip = two 16x16 WMMA
// accumulators sharing every B-fragment (halves hot-loop VMEM to 1 dword
// load per v_wmma_f32_16x16x4_f32). x tile (128x300) staged in LDS.
// A frag (16x4 f32): lane l (m=l&15, half=l>>4): a = {A[m][k0+2h], A[m][k0+2h+1]}
// B frag (4x16 f32): lane l (n=l&15):            b = {B[k0+2h][n], B[k0+2h+1][n]}
// C frag (16x16 f32): lane l, vgpr v -> C[v + 8*half][n=l&15]
// ---------------------------------------------------------------------------
__global__ __launch_bounds__(GATE_THREADS)
void grea_gate_wmma(const float* __restrict__ x, const float* __restrict__ w1p,
                    const float* __restrict__ gk, const float* __restrict__ gc,
                    const float* __restrict__ gw2, const float* __restrict__ gb2,
                    float* __restrict__ gate_out)
{
    extern __shared__ float xs[]; // MTILE * D_EMB floats = 150 KB
    const int tid = threadIdx.x;
    const int node0 = blockIdx.x * MTILE;

    // stage x tile (contiguous rows) as float4, fully coalesced
    const float4* xg = (const float4*)(x + (size_t)node0 * D_EMB);
    float4* xs4 = (float4*)xs;
    for (int i = tid; i < MTILE * D_EMB / 4; i += GATE_THREADS) xs4[i] = xg[i];
    __syncthreads();

    const int wave = tid >> 5;   // 0..3
    const int lane = tid & 31;
    const int half = lane >> 4;
    const int ln   = lane & 15;
    const int mrow = wave * 32;  // 32-row strip per wave

    const float* aptr0 = xs + (mrow + ln) * D_EMB + 2 * half;
    const float* aptr1 = xs + (mrow + 16 + ln) * D_EMB + 2 * half;

    float racc0[8], racc1[8];
#pragma unroll
    for (int v = 0; v < 8; ++v) { racc0[v] = 0.f; racc1[v] = 0.f; }

    for (int nt = 0; nt < HP / 16; ++nt) {
        const int n0 = nt * 16;
        const float* bptr = w1p + (size_t)(2 * half) * HP + n0 + ln;
        v8f c0 = {}, c1 = {};
#pragma unroll 5
        for (int k0 = 0; k0 < D_EMB; k0 += 4) {
            v2f b;
            b.x = bptr[(size_t)k0 * HP];
            b.y = bptr[(size_t)k0 * HP + HP];
            v2f a0 = *(const v2f*)(aptr0 + k0);
            v2f a1 = *(const v2f*)(aptr1 + k0);
            c0 = __builtin_amdgcn_wmma_f32_16x16x4_f32(false, a0, false, b,
                                                       (short)0, c0, false, false);
            c1 = __builtin_amdgcn_wmma_f32_16x16x4_f32(false, a1, false, b,
                                                       (short)0, c1, false, false);
        }
        // fused affine(BN) + ReLU + dot with w2 on the accumulator tiles
        const int h = n0 + ln;
        const float kk = gk[h], cc = gc[h], w2v = gw2[h];
#pragma unroll
        for (int v = 0; v < 8; ++v) {
            float z0 = c0[v] * kk + cc;
            racc0[v] += fmaxf(z0, 0.f) * w2v;
            float z1 = c1[v] * kk + cc;
            racc1[v] += fmaxf(z1, 0.f) * w2v;
        }
    }
    // reduce across the 16 lanes of each half-wave (sums over columns n)
#pragma unroll
    for (int v = 0; v < 8; ++v) {
        float s0 = racc0[v];
        s0 += __shfl_xor(s0, 1, 32);
        s0 += __shfl_xor(s0, 2, 32);
        s0 += __shfl_xor(s0, 4, 32);
        s0 += __shfl_xor(s0, 8, 32);
        racc0[v] = s0;
        float s1 = racc1[v];
        s1 += __shfl_xor(s1, 1, 32);
        s1 += __shfl_xor(s1, 2, 32);
        s1 += __shfl_xor(s1, 4, 32);
        s1 += __shfl_xor(s1, 8, 32);
        racc1[v] = s1;
    }
    if (ln == 0) {  // lanes 0 (rows +0..7) and 16 (rows +8..15) write results
        const float bias2 = gb2[0];
        const int mbase0 = node0 + mrow + 8 * half;
        const int mbase1 = mbase0 + 16;
#pragma unroll
        for (int v = 0; v < 8; ++v) {
            float g0 = 1.f / (1.f + __expf(-(racc0[v] + bias2)));
            gate_out[mbase0 + v] = g0;
            float g1 = 1.f / (1.f + __expf(-(racc1[v] + bias2)));
            gate_out[mbase1 + v] = g1;
        }
    }
}

// ---------------------------------------------------------------------------
// Kernel 3: segment sums per graph (batch is sorted -> binary search bounds)
// ---------------------------------------------------------------------------
__global__ __launch_bounds__(320)
void grea_segsum(const float* __restrict__ h_node, const float* __restrict__ gate,
                 const int* __restrict__ batch,
                 float* __restrict__ h_r, float* __restrict__ h_env)
{
    const int g = blockIdx.x;
    const int d = threadIdx.x;
    int lo = 0, hi = N_NODES;
    while (lo < hi) { int mid = (lo + hi) >> 1; if (batch[mid] < g) lo = mid + 1; else hi = mid; }
    const int start = lo;
    hi = N_NODES;
    while (lo < hi) { int mid = (lo + hi) >> 1; if (batch[mid] < g + 1) lo = mid + 1; else hi = mid; }
    const int end = lo;
    if (d >= D_EMB) return;
    float ar = 0.f, ae = 0.f;
    for (int n = start; n < end; ++n) {
        float gv = gate[n];
        float hv = h_node[(size_t)n * D_EMB + d];
        ar += gv * hv;
        ae += (1.f - gv) * hv;
    }
    h_r[(size_t)g * D_EMB + d] = ar;
    h_env[(size_t)g * D_EMB + d] = ae;
}

// ---------------------------------------------------------------------------
// Kernel 4: A2 = (h_r @ pred_w1) * kp ; B2 = (h_env @ pred_w1) * kp + cB
// (decomposition of the G^2-row predictor GEMM into two 512-row GEMMs)
// ---------------------------------------------------------------------------
__global__ __launch_bounds__(256)
void grea_proj(const float* __restrict__ h_r, const float* __restrict__ h_env,
               const float* __restrict__ pw1, const float* __restrict__ kp,
               const float* __restrict__ cB,
               float* __restrict__ A2, float* __restrict__ B2)
{
    int idx = blockIdx.x * 256 + threadIdx.x;
    if (idx >= 2 * G_GRAPHS * H_HID) return;
    int h = idx % H_HID;
    int g = (idx / H_HID) % G_GRAPHS;
    int mat = idx / (H_HID * G_GRAPHS);
    const float* src = (mat ? h_env : h_r) + (size_t)g * D_EMB;
    float acc = 0.f;
    for (int d = 0; d < D_EMB; ++d)
        acc = fmaf(src[d], pw1[(size_t)d * H_HID + h], acc);
    if (mat == 0) A2[(size_t)g * H_HID + h] = acc * kp[h];
    else          B2[(size_t)g * H_HID + h] = acc * kp[h] + cB[h];
}

// ---------------------------------------------------------------------------
// Kernel 5: per-i pairwise relu-dot, unbiased variance over 512 values, and
// prediction[i] = sum_h relu(A2[i,h] + cB[h]) * w2[h] + b2
// ---------------------------------------------------------------------------
__global__ __launch_bounds__(256)
void grea_pairwise(const float* __restrict__ A2, const float* __restrict__ B2,
                   const float* __restrict__ cB, const float* __restrict__ pw2,
                   const float* __restrict__ pb2,
                   float* __restrict__ out_pred, float* __restrict__ out_var)
{
    __shared__ float sA[H_HID], sW[H_HID], sC[H_HID];
    __shared__ float red[256], red2[256];
    const int i = blockIdx.x;
    const int tid = threadIdx.x;
    for (int h = tid; h < H_HID; h += 256) {
        sA[h] = A2[(size_t)i * H_HID + h];
        sW[h] = pw2[h];
        sC[h] = cB[h];
    }
    __syncthreads();
    const float b2 = pb2[0];
    const float* bp1 = B2 + (size_t)tid * H_HID;
    const float* bp2 = B2 + (size_t)(tid + 256) * H_HID;
    float acc1 = 0.f, acc2 = 0.f;
    for (int h = 0; h < H_HID; ++h) {
        float a = sA[h], w = sW[h];
        acc1 += fmaxf(a + bp1[h], 0.f) * w;
        acc2 += fmaxf(a + bp2[h], 0.f) * w;
    }
    float v1 = acc1 + b2, v2 = acc2 + b2;
    red[tid]  = v1 + v2;
    red2[tid] = v1 * v1 + v2 * v2;
    __syncthreads();
    for (int s = 128; s > 0; s >>= 1) {
        if (tid < s) { red[tid] += red[tid + s]; red2[tid] += red2[tid + s]; }
        __syncthreads();
    }
    if (tid == 0) {
        float S = red[0], Q = red2[0];
        out_var[i] = (Q - S * S * (1.0f / 512.0f)) * (1.0f / 511.0f);
    }
    __syncthreads();
    float p = 0.f;
    for (int h = tid; h < H_HID; h += 256)
        p += fmaxf(sA[h] + sC[h], 0.f) * sW[h];
    red[tid] = p;
    __syncthreads();
    for (int s = 128; s > 0; s >>= 1) {
        if (tid < s) red[tid] += red[tid + s];
        __syncthreads();
    }
    if (tid == 0) out_pred[i] = red[0] + b2;
}

// ---------------------------------------------------------------------------
extern "C" void kernel_launch(void* const* d_in, const int* in_sizes, int n_in,
                              void* d_out, int out_size, void* d_ws, size_t ws_size,
                              hipStream_t stream)
{
    (void)in_sizes; (void)n_in; (void)out_size; (void)ws_size;
    const float* h_node = (const float*)d_in[0];
    const float* x      = (const float*)d_in[1];
    const int*   batch  = (const int*)d_in[2];
    const float* gw1 = (const float*)d_in[4];
    const float* gb1 = (const float*)d_in[5];
    const float* gsc = (const float*)d_in[6];
    const float* gbi = (const float*)d_in[7];
    const float* gme = (const float*)d_in[8];
    const float* gva = (const float*)d_in[9];
    const float* gw2 = (const float*)d_in[10];
    const float* gb2 = (const float*)d_in[11];
    const float* pw1 = (const float*)d_in[12];
    const float* pb1 = (const float*)d_in[13];
    const float* psc = (const float*)d_in[14];
    const float* pbi = (const float*)d_in[15];
    const float* pme = (const float*)d_in[16];
    const float* pva = (const float*)d_in[17];
    const float* pw2 = (const float*)d_in[18];
    const float* pb2 = (const float*)d_in[19];

    float* ws   = (float*)d_ws;
    float* w1p  = ws;                 // 300*608 = 182400
    float* gk   = w1p + D_EMB * HP;   // 608
    float* gc   = gk + HP;            // 608
    float* gw2p = gc + HP;            // 608
    float* kp   = gw2p + HP;          // 600
    float* cB   = kp + H_HID;         // 600
    float* h_r  = cB + H_HID;                       // 512*300
    float* h_env= h_r + G_GRAPHS * D_EMB;           // 512*300
    float* A2   = h_env + G_GRAPHS * D_EMB;         // 512*600
    float* B2   = A2 + G_GRAPHS * H_HID;            // 512*600

    float* out      = (float*)d_out;
    float* out_pred = out;            // [512]
    float* out_var  = out + 512;      // [512]
    float* out_gate = out + 1024;     // [131072]

    grea_setup_params<<<(HP + 255) / 256, 256, 0, stream>>>(
        gb1, gsc, gbi, gme, gva, gw2, pb1, psc, pbi, pme, pva, gk, gc, gw2p, kp, cB);
    grea_pad_w1<<<(D_EMB * HP + 255) / 256, 256, 0, stream>>>(gw1, w1p);
    grea_gate_wmma<<<N_NODES / MTILE, GATE_THREADS, MTILE * D_EMB * sizeof(float), stream>>>(
        x, w1p, gk, gc, gw2p, gb2, out_gate);
    grea_segsum<<<G_GRAPHS, 320, 0, stream>>>(h_node, out_gate, batch, h_r, h_env);
    grea_proj<<<(2 * G_GRAPHS * H_HID + 255) / 256, 256, 0, stream>>>(
        h_r, h_env, pw1, kp, cB, A2, B2);
    grea_pairwise<<<G_GRAPHS, 256, 0, stream>>>(A2, B2, cB, pw2, pb2, out_pred, out_var);
}